// GNN_61469571940700
// MI455X (gfx1250) — compile-verified
//
#include <hip/hip_runtime.h>
#include <hip/hip_bf16.h>

typedef __attribute__((ext_vector_type(2))) float v2f;
typedef __attribute__((ext_vector_type(8))) float v8f;

#define N_NODES  50000
#define N_EDGES  800000
#define N_GRAPHS 512

// ---------- helpers ----------
__device__ __forceinline__ unsigned int fenc(float f) {
    unsigned int u = __float_as_uint(f);
    return u ^ ((u >> 31) ? 0xFFFFFFFFu : 0x80000000u);
}
__device__ __forceinline__ float fdec(unsigned int u) {
    unsigned int bits = (u & 0x80000000u) ? (u ^ 0x80000000u) : ~u;
    return __uint_as_float(bits);
}

// ---------- zero-pad layer-0 inputs so GEMM needs no K guards ----------
__global__ __launch_bounds__(256)
void pad_x_kernel(const float* __restrict__ x, float* __restrict__ xpad, int total /*N*12*/)
{
    int i = blockIdx.x * blockDim.x + threadIdx.x;
    if (i >= total) return;
    const int row = i / 12, k = i % 12;
    xpad[i] = (k < 9) ? x[row * 9 + k] : 0.0f;
}

__global__ __launch_bounds__(256)
void pad_w_kernel(const float* __restrict__ Wl0, const float* __restrict__ Wr0,
                  float* __restrict__ wpad /*[2][12][64]*/)
{
    int i = blockIdx.x * blockDim.x + threadIdx.x;
    if (i >= 2 * 12 * 64) return;
    const int m = i / 768, r = (i % 768) / 64, c = i % 64;
    const float* Ws = m ? Wr0 : Wl0;
    wpad[i] = (r < 9) ? Ws[r * 64 + c] : 0.0f;
}

// ---------- WMMA GEMM: xl = A@Wl + bl ; (DUAL) xr = A@Wr + br ----------
// A: [N, lda] row-major (lda == K), W: [K,64] row-major, fully unrolled K loop.
// One wave computes a 16-row x 64-col tile with V_WMMA_F32_16X16X4_F32.
template<int K, bool DUAL>
__global__ __launch_bounds__(32)
void gemm_kernel(const float* __restrict__ A, int lda,
                 const float* __restrict__ Wl, const float* __restrict__ Wr,
                 const float* __restrict__ bl, const float* __restrict__ br,
                 float* __restrict__ xl, float* __restrict__ xr)
{
    const int lane = threadIdx.x & 31;
    const int m0 = blockIdx.x * 16;
    const int lh = lane >> 4;        // 0 -> K pair (0,1), 1 -> K pair (2,3)
    const int lm = lane & 15;
    const int row = m0 + lm;

    v8f cl[4] = {};
    v8f cr[4] = {};

    #pragma unroll
    for (int k = 0; k < K; k += 4) {
        const int ka = k + 2 * lh;
        const v2f a = *(const v2f*)(A + (size_t)row * lda + ka);
        #pragma unroll
        for (int t = 0; t < 4; ++t) {
            const int col = t * 16 + lm;
            v2f b;
            b.x = Wl[ka * 64 + col];
            b.y = Wl[(ka + 1) * 64 + col];
            cl[t] = __builtin_amdgcn_wmma_f32_16x16x4_f32(false, a, false, b,
                                                          (short)0, cl[t], false, false);
            if (DUAL) {
                v2f b2;
                b2.x = Wr[ka * 64 + col];
                b2.y = Wr[(ka + 1) * 64 + col];
                cr[t] = __builtin_amdgcn_wmma_f32_16x16x4_f32(false, a, false, b2,
                                                              (short)0, cr[t], false, false);
            }
        }
    }

    // C/D layout: element i -> row m0 + i + 8*lh, col t*16 + lm
    #pragma unroll
    for (int t = 0; t < 4; ++t) {
        const int col = t * 16 + lm;
        const float bli = bl[col];
        #pragma unroll
        for (int i = 0; i < 8; ++i)
            xl[(size_t)(m0 + i + 8 * lh) * 64 + col] = cl[t][i] + bli;
        if (DUAL) {
            const float bri = br[col];
            #pragma unroll
            for (int i = 0; i < 8; ++i)
                xr[(size_t)(m0 + i + 8 * lh) * 64 + col] = cr[t][i] + bri;
        }
    }
}

// ---------- init segment-max buffer to enc(-inf) ----------
__global__ __launch_bounds__(256)
void init_menc_kernel(unsigned int* __restrict__ mE, int n)
{
    int i = blockIdx.x * blockDim.x + threadIdx.x;
    if (i < n) mE[i] = 0x007FFFFFu; // fenc(-inf)
}

// ---------- edge pass A: logits + segment max (wave per edge) ----------
__global__ __launch_bounds__(256)
void edge_logit_kernel(const float* __restrict__ xl, const float* __restrict__ xr,
                       const float* __restrict__ ea, const float* __restrict__ We,
                       const float* __restrict__ att,
                       const int* __restrict__ src, const int* __restrict__ dst,
                       float* __restrict__ logit, unsigned int* __restrict__ mE, int nE)
{
    const int e = blockIdx.x * (blockDim.x >> 5) + (threadIdx.x >> 5);
    const int lane = threadIdx.x & 31;
    if (e >= nE) return;

    // lane-0 loads scalars, broadcast
    int sd = 0;
    if (lane < 2) sd = src[e + lane * nE - lane * 0 + (lane ? (dst - src) - e + e : 0)];
    // (avoid cleverness: straightforward loads below)
    int s = 0, d = 0;
    if (lane == 0) { s = src[e]; d = dst[e]; }
    s = __shfl(s, 0, 32);
    d = __shfl(d, 0, 32);
    float eav = (lane < 3) ? ea[e * 3 + lane] : 0.0f;
    const float e0 = __shfl(eav, 0, 32);
    const float e1 = __shfl(eav, 1, 32);
    const float e2 = __shfl(eav, 2, 32);
    (void)sd;

    const int c0 = lane * 2;
    const float2 vl = *(const float2*)(xl + (size_t)s * 64 + c0);
    const float2 vr = *(const float2*)(xr + (size_t)d * 64 + c0);

    float sx = vl.x + vr.x + e0 * We[c0]     + e1 * We[64 + c0]     + e2 * We[128 + c0];
    float sy = vl.y + vr.y + e0 * We[c0 + 1] + e1 * We[64 + c0 + 1] + e2 * We[128 + c0 + 1];
    sx = sx > 0.0f ? sx : 0.2f * sx;   // LeakyReLU
    sy = sy > 0.0f ? sy : 0.2f * sy;
    float part = sx * att[c0] + sy * att[c0 + 1];
    #pragma unroll
    for (int off = 16; off > 0; off >>= 1) part += __shfl_xor(part, off, 32);
    if (lane == 0) {
        logit[e] = part;
        atomicMax(mE + d, fenc(part));
    }
}

// ---------- decode max + finite fix ----------
__global__ __launch_bounds__(256)
void fix_m_kernel(const unsigned int* __restrict__ mE, float* __restrict__ m, int n)
{
    int i = blockIdx.x * blockDim.x + threadIdx.x;
    if (i >= n) return;
    float f = fdec(mE[i]);
    if (!(f >= -3.0e38f && f <= 3.0e38f)) f = 0.0f;  // NaN/inf -> 0 (matches reference)
    m[i] = f;
}

// ---------- edge pass C: p = exp(logit - m[dst]); z[dst] += p ----------
__global__ __launch_bounds__(256)
void edge_softmax_kernel(const float* __restrict__ logit, const float* __restrict__ m,
                         const int* __restrict__ dst,
                         float* __restrict__ p, float* __restrict__ z, int nE)
{
    int e = blockIdx.x * blockDim.x + threadIdx.x;
    if (e >= nE) return;
    const int d = dst[e];
    const float pe = __expf(logit[e] - m[d]);
    p[e] = pe;
    atomicAdd(z + d, pe);
}

// ---------- edge pass D: acc[dst] += alpha * xl[src] (wave per edge) ----------
__global__ __launch_bounds__(256)
void edge_scatter_kernel(const float* __restrict__ xl, const float* __restrict__ p,
                         const float* __restrict__ z,
                         const int* __restrict__ src, const int* __restrict__ dst,
                         float* __restrict__ acc, int nE)
{
    const int e = blockIdx.x * (blockDim.x >> 5) + (threadIdx.x >> 5);
    const int lane = threadIdx.x & 31;
    if (e >= nE) return;

    int s = 0, d = 0;
    float alpha = 0.0f;
    if (lane == 0) {
        s = src[e];
        d = dst[e];
        alpha = p[e] / fmaxf(z[d], 1e-16f);
    }
    s = __shfl(s, 0, 32);
    d = __shfl(d, 0, 32);
    alpha = __shfl(alpha, 0, 32);

    const int c0 = lane * 2;
    const float2 vl = *(const float2*)(xl + (size_t)s * 64 + c0);
    atomicAdd(acc + (size_t)d * 64 + c0,     alpha * vl.x);
    atomicAdd(acc + (size_t)d * 64 + c0 + 1, alpha * vl.y);
}

// ---------- bias + BatchNorm(eval) + ReLU ----------
__global__ __launch_bounds__(256)
void bn_relu_kernel(const float* __restrict__ acc, const float* __restrict__ bo,
                    const float* __restrict__ g, const float* __restrict__ b,
                    const float* __restrict__ mm, const float* __restrict__ vv,
                    float* __restrict__ h, int total)
{
    int i = blockIdx.x * blockDim.x + threadIdx.x;
    if (i >= total) return;
    const int c = i & 63;
    float val = acc[i] + bo[c];
    val = (val - mm[c]) * (g[c] * rsqrtf(vv[c] + 1e-5f)) + b[c];
    h[i] = fmaxf(val, 0.0f);
}

// ---------- global mean pool: accumulate sums + counts (wave per node) ----------
__global__ __launch_bounds__(256)
void pool_kernel(const float* __restrict__ hjk, const int* __restrict__ batch,
                 float* __restrict__ pooled, float* __restrict__ cnt, int n)
{
    const int node = blockIdx.x * (blockDim.x >> 5) + (threadIdx.x >> 5);
    const int lane = threadIdx.x & 31;
    if (node >= n) return;
    int bg = 0;
    if (lane == 0) bg = batch[node];
    bg = __shfl(bg, 0, 32);
    const int c0 = lane * 2;
    const float2 v = *(const float2*)(hjk + (size_t)node * 64 + c0);
    atomicAdd(pooled + (size_t)bg * 64 + c0,     v.x);
    atomicAdd(pooled + (size_t)bg * 64 + c0 + 1, v.y);
    if (lane == 0) atomicAdd(cnt + bg, 1.0f);
}

// ---------- head: out[g] = (pooled[g]/cnt) . Whead + bhead (wave per graph) ----------
__global__ __launch_bounds__(256)
void head_kernel(const float* __restrict__ pooled, const float* __restrict__ cnt,
                 const float* __restrict__ Whead, const float* __restrict__ bhead,
                 float* __restrict__ out, int nG)
{
    const int g = blockIdx.x * (blockDim.x >> 5) + (threadIdx.x >> 5);
    const int lane = threadIdx.x & 31;
    if (g >= nG) return;
    const float invc = 1.0f / fmaxf(cnt[g], 1.0f);
    const int c0 = lane * 2;
    float part = pooled[(size_t)g * 64 + c0]     * invc * Whead[c0]
               + pooled[(size_t)g * 64 + c0 + 1] * invc * Whead[c0 + 1];
    #pragma unroll
    for (int off = 16; off > 0; off >>= 1) part += __shfl_xor(part, off, 32);
    if (lane == 0) out[g] = part + bhead[0];
}

extern "C" void kernel_launch(void* const* d_in, const int* in_sizes, int n_in,
                              void* d_out, int out_size, void* d_ws, size_t ws_size,
                              hipStream_t stream)
{
    const int N = N_NODES, E = N_EDGES, G = N_GRAPHS;

    const float* x     = (const float*)d_in[0];
    const float* ea    = (const float*)d_in[1];
    const float* Wl0   = (const float*)d_in[2];
    const float* Wr0   = (const float*)d_in[3];
    const float* bl0   = (const float*)d_in[4];
    const float* br0   = (const float*)d_in[5];
    const float* We0   = (const float*)d_in[6];
    const float* att0  = (const float*)d_in[7];
    const float* bo0   = (const float*)d_in[8];
    const float* Wl    = (const float*)d_in[9];
    const float* Wr    = (const float*)d_in[10];
    const float* bl    = (const float*)d_in[11];
    const float* br    = (const float*)d_in[12];
    const float* We    = (const float*)d_in[13];
    const float* att   = (const float*)d_in[14];
    const float* bo    = (const float*)d_in[15];
    const float* bn_g  = (const float*)d_in[16];
    const float* bn_b  = (const float*)d_in[17];
    const float* bn_m  = (const float*)d_in[18];
    const float* bn_v  = (const float*)d_in[19];
    const float* Wjk   = (const float*)d_in[20];
    const float* bjk   = (const float*)d_in[21];
    const float* Whead = (const float*)d_in[22];
    const float* bhead = (const float*)d_in[23];
    const int*   eidx  = (const int*)d_in[24];
    const int*   batch = (const int*)d_in[25];

    const int* srcI = eidx;        // edge_index[0]
    const int* dstI = eidx + E;    // edge_index[1]

    // workspace layout (floats unless noted)
    float* ws      = (float*)d_ws;
    float* h       = ws;                      // N*64
    float* xl      = h + (size_t)N * 64;      // N*64
    float* xr      = xl + (size_t)N * 64;     // N*64
    float* acc     = xr + (size_t)N * 64;     // N*64
    float* logit   = acc + (size_t)N * 64;    // E
    float* p       = logit + E;               // E
    float* mF      = p + E;                   // N
    float* z       = mF + N;                  // N
    unsigned* mE   = (unsigned*)(z + N);      // N (u32)
    float* pooled  = (float*)(mE + N);        // G*64
    float* cnt     = pooled + (size_t)G * 64; // G
    float* xpad    = cnt + G;                 // N*12
    float* wpad    = xpad + (size_t)N * 12;   // 2*12*64

    const int gemmBlocks = N / 16;              // 3125
    const int edgeWaveBlocks = (E + 7) / 8;     // wave per edge, 8 waves/block
    const int nodeWaveBlocks = (N + 7) / 8;

    // one-time padding of layer-0 operands (deterministic; re-done every call)
    pad_x_kernel<<<(N * 12 + 255) / 256, 256, 0, stream>>>(x, xpad, N * 12);
    pad_w_kernel<<<(1536 + 255) / 256, 256, 0, stream>>>(Wl0, Wr0, wpad);

    for (int L = 0; L < 5; ++L) {
        const float *We_, *att_, *bo_;
        if (L == 0) {
            We_ = We0; att_ = att0; bo_ = bo0;
            gemm_kernel<12, true><<<gemmBlocks, 32, 0, stream>>>(
                xpad, 12, wpad, wpad + 768, bl0, br0, xl, xr);
        } else {
            const int i = L - 1;
            We_ = We + (size_t)i * 3 * 64; att_ = att + i * 64; bo_ = bo + i * 64;
            gemm_kernel<64, true><<<gemmBlocks, 32, 0, stream>>>(
                h, 64,
                Wl + (size_t)i * 64 * 64, Wr + (size_t)i * 64 * 64,
                bl + i * 64, br + i * 64, xl, xr);
        }

        hipMemsetAsync(acc, 0, (size_t)N * 64 * sizeof(float), stream);
        hipMemsetAsync(z, 0, (size_t)N * sizeof(float), stream);
        init_menc_kernel<<<(N + 255) / 256, 256, 0, stream>>>(mE, N);

        edge_logit_kernel<<<edgeWaveBlocks, 256, 0, stream>>>(
            xl, xr, ea, We_, att_, srcI, dstI, logit, mE, E);
        fix_m_kernel<<<(N + 255) / 256, 256, 0, stream>>>(mE, mF, N);
        edge_softmax_kernel<<<(E + 255) / 256, 256, 0, stream>>>(logit, mF, dstI, p, z, E);
        edge_scatter_kernel<<<edgeWaveBlocks, 256, 0, stream>>>(xl, p, z, srcI, dstI, acc, E);

        bn_relu_kernel<<<(N * 64 + 255) / 256, 256, 0, stream>>>(
            acc, bo_, bn_g + L * 64, bn_b + L * 64, bn_m + L * 64, bn_v + L * 64,
            h, N * 64);
    }

    // jk='last' projection: hjk = h @ Wjk + bjk  (stored in xl)
    gemm_kernel<64, false><<<gemmBlocks, 32, 0, stream>>>(
        h, 64, Wjk, nullptr, bjk, nullptr, xl, nullptr);

    // global mean pool + head
    hipMemsetAsync(pooled, 0, (size_t)(G * 64 + G) * sizeof(float), stream);
    pool_kernel<<<nodeWaveBlocks, 256, 0, stream>>>(xl, batch, pooled, cnt, N);
    head_kernel<<<(G + 7) / 8, 256, 0, stream>>>(pooled, cnt, Whead, bhead,
                                                 (float*)d_out, G);
}